// QNearestNeighbourEuclidean_11819749998731
// MI455X (gfx1250) — compile-verified
//
#include <hip/hip_runtime.h>

// Problem constants from the reference
#define NB 8
#define NV 2048
#define NS 4
#define NF 64
#define NK 8
#define TILE 16
#define NT (NV / TILE)          // 128 v-tiles per batch
#define KOUT (NK - 1)           // 7 neighbours kept
#define CHUNK 512               // vertices per async staging chunk (8KB)

#define MAXD 3.4028234663852886e38f   // FLT_MAX (reference MAX_DIST)

typedef __attribute__((ext_vector_type(2))) float v2f;
typedef __attribute__((ext_vector_type(8))) float v8f;
typedef __attribute__((ext_vector_type(4))) int   v4i;
typedef unsigned long long u64;

// pointer types matching the async-to-LDS builtin's expected parameters
typedef __attribute__((address_space(1))) v4i* gptr4;   // global (device) int4*
typedef __attribute__((address_space(3))) v4i* lptr4;   // LDS int4*

#if __has_builtin(__builtin_amdgcn_global_load_async_to_lds_b128) && \
    __has_builtin(__builtin_amdgcn_s_wait_asynccnt)
#define HAS_ASYNC_LDS 1
#else
#define HAS_ASYNC_LDS 0
#endif

__global__ __launch_bounds__(32)
void knn_wmma_kernel(const float* __restrict__ coords,   // [B,V,S]
                     const float* __restrict__ feats,    // [B,V,F]
                     const float* __restrict__ active,   // [B,1]
                     float* __restrict__ out_dist,       // [B,V,7]
                     float* __restrict__ out_feat)       // [B,V,7,F]
{
    __shared__ float s_norm[NV];            // |c_w|^2 for all vertices of this batch
    __shared__ u64   s_mkey[TILE][2][9];    // per-column, per-half sorted top-8 (+sentinel)
    __shared__ int   s_nb[TILE][KOUT];      // final neighbour indices per row
#if HAS_ASYNC_LDS
    __shared__ float s_cbuf[CHUNK * NS];    // async staging buffer (8KB)
#endif

    const int lane  = threadIdx.x;      // 0..31 (wave32)
    const int b     = blockIdx.x / NT;
    const int vtile = blockIdx.x % NT;
    const int vbase = vtile * TILE;
    const int col   = lane & 15;        // column (v) this lane owns in D
    const int half  = lane >> 4;        // K-pair for A/B, row group for D

    const float af = active[b];
    // valid w  <=>  (float)w < af  <=>  w < ceil(af)  (exact for integer af too)
    const int wlimit = (int)ceilf(fminf(fmaxf(af, 0.0f), (float)NV));
    const int ntiles = (wlimit + TILE - 1) / TILE;

    const float* cb = coords + (size_t)b * NV * NS;

    // ---- Phase 0: all vertex norms into LDS ----
#if HAS_ASYNC_LDS
    for (int c0 = 0; c0 < NV; c0 += CHUNK) {
        // each lane async-copies its own 16B rows (one vertex = one b128)
        for (int i = lane; i < CHUNK; i += 32) {
            __builtin_amdgcn_global_load_async_to_lds_b128(
                (gptr4)(cb + (size_t)(c0 + i) * NS),
                (lptr4)&s_cbuf[i * NS],
                0, 0);
        }
        __builtin_amdgcn_s_wait_asynccnt(0);
        for (int i = lane; i < CHUNK; i += 32) {
            float4 c4 = *(const float4*)&s_cbuf[i * NS];
            s_norm[c0 + i] = c4.x * c4.x + c4.y * c4.y + c4.z * c4.z + c4.w * c4.w;
        }
        // ds_load results are consumed above (compiler waits DScnt) before reuse
    }
#else
    for (int i = lane; i < NV; i += 32) {
        float4 c4 = *(const float4*)(cb + i * NS);
        s_norm[i] = c4.x * c4.x + c4.y * c4.y + c4.z * c4.z + c4.w * c4.w;
    }
#endif
    __syncthreads();

    // ---- B matrix: this block's v-tile coordinates (fixed across the loop) ----
    // B[k][n] = coord[vbase+n][k]; lane holds N=col, K = {2*half, 2*half+1}
    const int vcol = vbase + col;
    v2f bm;
    bm.x = cb[vcol * NS + 2 * half + 0];
    bm.y = cb[vcol * NS + 2 * half + 1];
    const float nv = s_norm[vcol];

    // ---- per-lane running top-8 as packed u64 keys: (dist_bits<<32)|index ----
    // Non-negative float bits are order-isomorphic to u32, so one u64 compare
    // gives the exact (value, lowest-index) ordering of jax.lax.top_k.
    u64 key[NK];
#pragma unroll
    for (int k = 0; k < NK; ++k) key[k] = ~0ull;

    // ---- main loop over w tiles: one V_WMMA_F32_16X16X4_F32 per tile ----
    // Software-pipelined: fetch tile t+1's A coords + norms during tile t's
    // WMMA + insert work so the loads' latency is covered by real work.
    v2f    aN;                   // next tile A coords (unscaled)
    float4 n0N, n1N;             // next tile norms
    {
        const int wrow = col;            // tile 0
        aN.x = cb[wrow * NS + 2 * half + 0];
        aN.y = cb[wrow * NS + 2 * half + 1];
        const int wr0 = 8 * half;
        n0N = *(const float4*)&s_norm[wr0];
        n1N = *(const float4*)&s_norm[wr0 + 4];
    }

    for (int t = 0; t < ntiles; ++t) {
        const v2f    acur = aN;
        const float4 n0c  = n0N;
        const float4 n1c  = n1N;

        // prefetch next tile (re-fetch same tile on the last iteration)
        {
            const int tn   = (t + 1 < ntiles) ? (t + 1) : t;
            const int wrow = tn * TILE + col;
            aN.x = cb[wrow * NS + 2 * half + 0];
            aN.y = cb[wrow * NS + 2 * half + 1];
            const int wr0 = tn * TILE + 8 * half;
            n0N = *(const float4*)&s_norm[wr0];
            n1N = *(const float4*)&s_norm[wr0 + 4];
        }

        // A[m][k] = -2 * coord[wbase+m][k]; lane holds M=col, K = {2*half, 2*half+1}
        v2f am;
        am.x = -2.0f * acur.x;
        am.y = -2.0f * acur.y;

        v8f c = {0.f, 0.f, 0.f, 0.f, 0.f, 0.f, 0.f, 0.f};
        v8f d = __builtin_amdgcn_wmma_f32_16x16x4_f32(
            /*neg_a=*/false, am, /*neg_b=*/false, bm,
            /*c_mod=*/(short)0, c, /*reuse_a=*/false, /*reuse_b=*/false);
        // D[m][n] = -2 * c_w[m] . c_v[n]; this lane: n = col, m = 8*half + j

        const int wr0 = t * TILE + 8 * half;
        const float nw[8] = {n0c.x, n0c.y, n0c.z, n0c.w,
                             n1c.x, n1c.y, n1c.z, n1c.w};

#pragma unroll
        for (int j = 0; j < 8; ++j) {
            const int w = wr0 + j;
            float val = fabsf(d[j] + nw[j] + nv);
            if (w >= wlimit) val = MAXD;            // masked candidate
            u64 ck = ((u64)__float_as_uint(val) << 32) | (unsigned)w;
            if (ck < key[NK - 1]) {                 // execz-skip when no lane inserts
                // branchless bubble: v_min_u64/v_max_u64 compare-swap ladder
#pragma unroll
                for (int i = 0; i < NK; ++i) {
                    const bool sw = ck < key[i];
                    const u64 lo = sw ? ck : key[i];
                    const u64 hi = sw ? key[i] : ck;
                    key[i] = lo;
                    ck = hi;
                }
            }
        }
    }

    // ---- stash both halves' sorted key lists in LDS, sentinel in slot 8 ----
#pragma unroll
    for (int k = 0; k < NK; ++k) s_mkey[col][half][k] = key[k];
    s_mkey[col][half][8] = ~0ull;
    __syncthreads();

    // ---- lanes 0..15: merge the two sorted 8-lists (keys order lexicographically) ----
    if (lane < 16) {
        const int v = vbase + lane;
        u64 mk[NK];
        int ia = 0, ib = 0;
        u64 ka = s_mkey[lane][0][0];
        u64 kb = s_mkey[lane][1][0];
#pragma unroll
        for (int k = 0; k < NK; ++k) {
            const bool ta = ka < kb;
            mk[k] = ta ? ka : kb;
            if (ta) { ++ia; ka = s_mkey[lane][0][ia]; }
            else    { ++ib; kb = s_mkey[lane][1][ib]; }
        }

        const bool rowvalid = (float)v < af;   // fully-masked row => all MAX_DIST
        float* od = out_dist + ((size_t)b * NV + v) * KOUT;
#pragma unroll
        for (int k = 1; k < NK; ++k) {          // drop entry 0 (self)
            const float dv = rowvalid ? __uint_as_float((unsigned)(mk[k] >> 32)) : MAXD;
            const int   iv = rowvalid ? (int)(unsigned)(mk[k] & 0xffffffffu) : k;
            od[k - 1]         = dv;
            s_nb[lane][k - 1] = iv;
        }
    }
    __syncthreads();

    // ---- feature gather: 16 rows x 7 neighbours x 64 floats (256B each) ----
    const float* fb = feats + (size_t)b * NV * NF;
    float* of = out_feat + ((size_t)b * NV + vbase) * KOUT * NF;
    for (int row = 0; row < TILE; ++row) {
#pragma unroll
        for (int nb = 0; nb < KOUT; ++nb) {
            const int w = s_nb[row][nb];
            const float2 src = *(const float2*)(fb + (size_t)w * NF + lane * 2);
            *(float2*)(of + ((size_t)row * KOUT + nb) * NF + lane * 2) = src;
        }
    }
}

extern "C" void kernel_launch(void* const* d_in, const int* in_sizes, int n_in,
                              void* d_out, int out_size, void* d_ws, size_t ws_size,
                              hipStream_t stream) {
    const float* coords = (const float*)d_in[0];   // (B,V,S) f32
    const float* feats  = (const float*)d_in[1];   // (B,V,F) f32
    const float* active = (const float*)d_in[2];   // (B,1)   f32

    float* out_dist = (float*)d_out;                       // (B,V,7)
    float* out_feat = out_dist + (size_t)NB * NV * KOUT;   // (B,V,7,F)

    dim3 grid(NB * NT);   // 1024 blocks, one wave32 each (one 16-row v-tile)
    dim3 block(32);
    knn_wmma_kernel<<<grid, block, 0, stream>>>(coords, feats, active,
                                                out_dist, out_feat);
}